// WordConvLayer2_56581899157981
// MI455X (gfx1250) — compile-verified
//
#include <hip/hip_runtime.h>
#include <hip/hip_bf16.h>

// ---------------------------------------------------------------------------
// GNN layer: h_proj = h@W + b (fp32 WMMA); msg = h_proj[src]*ew; mean-agg by
// dst (fp32 global atomics); relu.  N=50000, E=200000, D=512.
// ---------------------------------------------------------------------------

#define D_DIM 512

typedef __attribute__((ext_vector_type(2))) float v2f;
typedef __attribute__((ext_vector_type(8))) float v8f;

// ---------------------------------------------------------------------------
// Zero-fill (float4 granularity). Replaces hipMemsetAsync to stay trivially
// graph-capture safe.
// ---------------------------------------------------------------------------
__global__ void gnn_zero_kernel(float4* __restrict__ p, long n4) {
    long i = (long)blockIdx.x * blockDim.x + threadIdx.x;
    if (i < n4) p[i] = make_float4(0.f, 0.f, 0.f, 0.f);
}

// ---------------------------------------------------------------------------
// GEMM: h_proj = h @ W + b  via V_WMMA_F32_16X16X4_F32.
// One wave computes a 16(M) x 64(N) strip: 4 accumulators, A fragment reused
// across the 4 N-subtiles. Grid is exact (M%16==0, N==512), so EXEC is all-1s
// for every WMMA as the ISA requires.
//
// Fragment layouts (ISA 7.12.2, fp32):
//   A 16x4 : lane l -> row M=(l&15); half-wave (l>>4) selects K-pair;
//            VGPR0=K+0, VGPR1=K+1  -> contiguous v2f load per lane.
//   B 4x16 : lane l -> col N=(l&15); half-wave selects K-pair (rows k, k+1).
//   C 16x16: lane l -> col N=(l&15); VGPR r -> row M = r + 8*(l>>4).
// ---------------------------------------------------------------------------
__global__ void gnn_gemm_wmma_kernel(const float* __restrict__ h,
                                     const float* __restrict__ W,
                                     const float* __restrict__ bias,
                                     float* __restrict__ hproj,
                                     int M) {
    const int lane  = threadIdx.x & 31;
    const int gwave = (blockIdx.x * blockDim.x + threadIdx.x) >> 5;

    const int tm = gwave >> 3;        // M tile index (16 rows each)
    const int ng = gwave & 7;         // N group index (64 cols each)
    if (tm * 16 >= M) return;         // exact grids keep this uniform

    const int l16   = lane & 15;
    const int khalf = lane >> 4;      // 0: K={0,1}, 1: K={2,3} within step
    const int koff  = khalf * 2;

    const float* Arow  = h + (size_t)(tm * 16 + l16) * D_DIM + koff;
    const int    nbase = ng * 64 + l16;

    v8f acc0 = {}, acc1 = {}, acc2 = {}, acc3 = {};

    #pragma unroll 4
    for (int k0 = 0; k0 < D_DIM; k0 += 4) {
        v2f a = *(const v2f*)(Arow + k0);                      // A[m][k0+koff .. +1]
        const float* Bk = W + (size_t)(k0 + koff) * D_DIM + nbase;
        v2f b0, b1, b2, b3;
        b0.x = Bk[0];          b0.y = Bk[0 + D_DIM];
        b1.x = Bk[16];         b1.y = Bk[16 + D_DIM];
        b2.x = Bk[32];         b2.y = Bk[32 + D_DIM];
        b3.x = Bk[48];         b3.y = Bk[48 + D_DIM];
        acc0 = __builtin_amdgcn_wmma_f32_16x16x4_f32(false, a, false, b0, (short)0, acc0, false, false);
        acc1 = __builtin_amdgcn_wmma_f32_16x16x4_f32(false, a, false, b1, (short)0, acc1, false, false);
        acc2 = __builtin_amdgcn_wmma_f32_16x16x4_f32(false, a, false, b2, (short)0, acc2, false, false);
        acc3 = __builtin_amdgcn_wmma_f32_16x16x4_f32(false, a, false, b3, (short)0, acc3, false, false);
    }

    const int mbase = tm * 16 + khalf * 8;
    v8f accs[4] = {acc0, acc1, acc2, acc3};
    #pragma unroll
    for (int j = 0; j < 4; ++j) {
        const int n  = nbase + j * 16;
        const float bn = bias[n];
        #pragma unroll
        for (int r = 0; r < 8; ++r) {
            hproj[(size_t)(mbase + r) * D_DIM + n] = accs[j][r] + bn;
        }
    }
}

// ---------------------------------------------------------------------------
// Edge scatter: one 128-thread block per edge. Coalesced float4 read of the
// source row (usually an L2 hit: h_proj is 102 MB < 192 MB L2), scaled by the
// edge weight, accumulated with native fp32 global atomics into d_out.
// Thread 0 bumps the in-degree counter.
// ---------------------------------------------------------------------------
__global__ void gnn_scatter_kernel(const float* __restrict__ hproj,
                                   const float* __restrict__ ew,
                                   const int* __restrict__ src,
                                   const int* __restrict__ dst,
                                   float* __restrict__ sums,
                                   float* __restrict__ cnt,
                                   int E) {
    const int e = blockIdx.x;
    if (e >= E) return;
    const int   s = src[e];
    const int   d = dst[e];
    const float w = ew[e];

    const float4* in  = (const float4*)(hproj + (size_t)s * D_DIM);
    float*        out = sums + (size_t)d * D_DIM;

    const int t = threadIdx.x;            // 128 threads * 4 floats = 512
    float4 v = in[t];
    float* o = out + 4 * t;
    __hip_atomic_fetch_add(o + 0, v.x * w, __ATOMIC_RELAXED, __HIP_MEMORY_SCOPE_AGENT);
    __hip_atomic_fetch_add(o + 1, v.y * w, __ATOMIC_RELAXED, __HIP_MEMORY_SCOPE_AGENT);
    __hip_atomic_fetch_add(o + 2, v.z * w, __ATOMIC_RELAXED, __HIP_MEMORY_SCOPE_AGENT);
    __hip_atomic_fetch_add(o + 3, v.w * w, __ATOMIC_RELAXED, __HIP_MEMORY_SCOPE_AGENT);
    if (t == 0)
        __hip_atomic_fetch_add(cnt + d, 1.0f, __ATOMIC_RELAXED, __HIP_MEMORY_SCOPE_AGENT);
}

// ---------------------------------------------------------------------------
// Finalize in place: out = relu(out / max(cnt,1)).
// ---------------------------------------------------------------------------
__global__ void gnn_finalize_kernel(float4* __restrict__ out,
                                    const float* __restrict__ cnt,
                                    long total4) {
    long i = (long)blockIdx.x * blockDim.x + threadIdx.x;
    if (i >= total4) return;
    const int node = (int)(i >> 7);           // 128 float4 per node (512 floats)
    const float inv = 1.0f / fmaxf(cnt[node], 1.0f);
    float4 v = out[i];
    v.x = fmaxf(v.x * inv, 0.0f);
    v.y = fmaxf(v.y * inv, 0.0f);
    v.z = fmaxf(v.z * inv, 0.0f);
    v.w = fmaxf(v.w * inv, 0.0f);
    out[i] = v;
}

// ---------------------------------------------------------------------------
// Launcher. Inputs (setup_inputs order):
//   0: h [N,512] f32   1: W [512,512] f32   2: b [512] f32
//   3: edge_weight [E] f32   4: src [E] i32   5: dst [E] i32
// d_out: [N,512] f32.  d_ws: h_proj (N*512*4 B) + cnt (N*4 B).
// ---------------------------------------------------------------------------
extern "C" void kernel_launch(void* const* d_in, const int* in_sizes, int n_in,
                              void* d_out, int out_size, void* d_ws, size_t ws_size,
                              hipStream_t stream) {
    const float* h  = (const float*)d_in[0];
    const float* W  = (const float*)d_in[1];
    const float* b  = (const float*)d_in[2];
    const float* ew = (const float*)d_in[3];
    const int* src  = (const int*)d_in[4];
    const int* dst  = (const int*)d_in[5];
    float* out      = (float*)d_out;

    const int N = in_sizes[0] / D_DIM;    // 50000 (multiple of 16)
    const int E = in_sizes[3];            // 200000

    char*  ws    = (char*)d_ws;
    float* hproj = (float*)ws;                                  // N*512 floats
    float* cnt   = (float*)(ws + (size_t)N * D_DIM * sizeof(float)); // N floats

    // 1) zero accumulators (d_out) and degree counts (cnt)
    {
        long n4 = (long)N * D_DIM / 4;
        gnn_zero_kernel<<<(unsigned)((n4 + 255) / 256), 256, 0, stream>>>((float4*)out, n4);
        long c4 = (N + 3) / 4;
        gnn_zero_kernel<<<(unsigned)((c4 + 255) / 256), 256, 0, stream>>>((float4*)cnt, c4);
    }

    // 2) h_proj = h @ W + b  (WMMA fp32). One wave per 16x64 strip.
    {
        const int waves  = (N / 16) * (D_DIM / 64);   // 3125 * 8 = 25000
        const int blocks = waves / 8;                 // 8 waves (256 thr) / block
        gnn_gemm_wmma_kernel<<<blocks, 256, 0, stream>>>(h, W, b, hproj, N);
    }

    // 3) edge scatter with fp32 atomics
    gnn_scatter_kernel<<<E, 128, 0, stream>>>(hproj, ew, src, dst, out, cnt, E);

    // 4) mean + relu in place
    {
        long n4 = (long)N * D_DIM / 4;
        gnn_finalize_kernel<<<(unsigned)((n4 + 255) / 256), 256, 0, stream>>>((float4*)out, cnt, n4);
    }
}